// MinRNNPredictor_13065290514804
// MI455X (gfx1250) — compile-verified
//
#include <hip/hip_runtime.h>
#include <hip/hip_bf16.h>
#include <math.h>

typedef __attribute__((ext_vector_type(2))) float v2f;
typedef __attribute__((ext_vector_type(4))) float v4f;
typedef __attribute__((ext_vector_type(8))) float v8f;

// Problem dims (from reference setup_inputs)
#define BSZ   8
#define TSEQ  4096
#define DIN   512
#define HDIM  1024
#define DOUT  512
#define MROWS (BSZ * TSEQ)     // 32768

// GEMM tiling
#define TM 128
#define TN 64
#define TK 32
#define TKP (TK + 2)           // even pad: 8B-aligned fragments, conflict-free

// Scan chunking
#define CHUNK   128
#define NCHUNK  (TSEQ / CHUNK) // 32

static __device__ __forceinline__ float sigmoidf_fast(float x) {
    return 1.0f / (1.0f + __expf(-x));
}

// ---------------------------------------------------------------------------
// Fused dual GEMM + minGRU gate epilogue:
//   accZ = X @ Wz ; accH = X @ Wh
//   z = sigmoid(accZ + bz); outA = 1 - z ; outB = z * (accH + bh)
// X: [M,K] row-major, W: [K,N] row-major, outputs [M,N].
// Block tile 128x64, 8 waves, each wave 32x32 (2x2 WMMA 16x16 tiles),
// K accumulated via V_WMMA_F32_16X16X4_F32.
// LDS: X tile [TM][TKP]; W tiles stored TRANSPOSED [TN][TKP] so each lane's
// B fragment (two consecutive K for one column) is one aligned ds_load_b64.
// ---------------------------------------------------------------------------
__global__ __launch_bounds__(256)
void gemm_dual_mingru(const float* __restrict__ X,
                      const float* __restrict__ Wz, const float* __restrict__ bz,
                      const float* __restrict__ Wh, const float* __restrict__ bh,
                      float* __restrict__ outA, float* __restrict__ outB,
                      int M, int K, int N)
{
    __shared__ float sX [TM][TKP];
    __shared__ float sWz[TN][TKP];   // transposed: [n][k]
    __shared__ float sWh[TN][TKP];

    const int tid   = threadIdx.x;
    const int lane  = tid & 31;
    const int wave  = tid >> 5;        // 0..7
    const int waveM = wave >> 1;       // 0..3  -> 32-row strip
    const int waveN = wave & 1;        // 0..1  -> 32-col strip
    const int bm = blockIdx.x * TM;
    const int bn = blockIdx.y * TN;

    v8f accZ[2][2] = {};
    v8f accH[2][2] = {};

    const int am  = lane & 15;         // A row within 16
    const int ak  = (lane >> 4) * 2;   // K sub-offset (0 or 2) per lane half
    const int bnl = lane & 15;         // B col within 16

    for (int k0 = 0; k0 < K; k0 += TK) {
        __syncthreads();
        // Stage X tile [TM x TK] with b128 global loads (4 float4 per thread)
        for (int j = tid; j < (TM * TK) / 4; j += 256) {
            int r  = j >> 3;            // TK/4 = 8 float4 per row
            int cq = (j & 7) * 4;
            v4f v = *(const v4f*)&X[(size_t)(bm + r) * K + (k0 + cq)];
            sX[r][cq + 0] = v.x; sX[r][cq + 1] = v.y;
            sX[r][cq + 2] = v.z; sX[r][cq + 3] = v.w;
        }
        // Stage W tiles [TK x TN] -> transposed LDS (2 float4 per thread each)
        for (int j = tid; j < (TK * TN) / 4; j += 256) {
            int r  = j >> 4;            // TN/4 = 16 float4 per row
            int cq = (j & 15) * 4;
            v4f vz = *(const v4f*)&Wz[(size_t)(k0 + r) * N + (bn + cq)];
            v4f vh = *(const v4f*)&Wh[(size_t)(k0 + r) * N + (bn + cq)];
            sWz[cq + 0][r] = vz.x; sWz[cq + 1][r] = vz.y;
            sWz[cq + 2][r] = vz.z; sWz[cq + 3][r] = vz.w;
            sWh[cq + 0][r] = vh.x; sWh[cq + 1][r] = vh.y;
            sWh[cq + 2][r] = vh.z; sWh[cq + 3][r] = vh.w;
        }
        __syncthreads();

#pragma unroll
        for (int kk = 0; kk < TK; kk += 4) {
            // A fragments (16x4 f32): lanes 0-15 hold K=kk..kk+1, lanes 16-31 K=kk+2..kk+3
            v2f afr[2];
#pragma unroll
            for (int i = 0; i < 2; ++i) {
                int row = waveM * 32 + i * 16 + am;
                afr[i] = *(const v2f*)&sX[row][kk + ak];        // ds_load_b64
            }
            // B fragments (4x16 f32): contiguous K pair from transposed tile
            v2f bzf[2], bhf[2];
#pragma unroll
            for (int j = 0; j < 2; ++j) {
                int col = waveN * 32 + j * 16 + bnl;
                bzf[j] = *(const v2f*)&sWz[col][kk + ak];       // ds_load_b64
                bhf[j] = *(const v2f*)&sWh[col][kk + ak];
            }
#pragma unroll
            for (int i = 0; i < 2; ++i)
#pragma unroll
                for (int j = 0; j < 2; ++j) {
                    accZ[i][j] = __builtin_amdgcn_wmma_f32_16x16x4_f32(
                        false, afr[i], false, bzf[j], (short)0, accZ[i][j], false, false);
                    accH[i][j] = __builtin_amdgcn_wmma_f32_16x16x4_f32(
                        false, afr[i], false, bhf[j], (short)0, accH[i][j], false, false);
                }
        }
    }

    // Epilogue: D layout — VGPR r: lanes 0-15 => (M=r, N=lane), lanes 16-31 => (M=r+8)
    const int col0  = lane & 15;
    const int rhalf = (lane >> 4) * 8;
#pragma unroll
    for (int i = 0; i < 2; ++i) {
#pragma unroll
        for (int j = 0; j < 2; ++j) {
            int ncol = bn + waveN * 32 + j * 16 + col0;
            float bzv = bz[ncol];
            float bhv = bh[ncol];
#pragma unroll
            for (int r = 0; r < 8; ++r) {
                int row = bm + waveM * 32 + i * 16 + r + rhalf;
                float z  = sigmoidf_fast(accZ[i][j][r] + bzv);
                float ht = accH[i][j][r] + bhv;
                size_t idx = (size_t)row * N + ncol;
                outA[idx] = 1.0f - z;
                outB[idx] = z * ht;
            }
        }
    }
}

// ---------------------------------------------------------------------------
// Single GEMM + bias (final FC): out = X @ W + b
// ---------------------------------------------------------------------------
__global__ __launch_bounds__(256)
void gemm_bias(const float* __restrict__ X,
               const float* __restrict__ W, const float* __restrict__ bias,
               float* __restrict__ out, int M, int K, int N)
{
    __shared__ float sX[TM][TKP];
    __shared__ float sW[TN][TKP];    // transposed

    const int tid   = threadIdx.x;
    const int lane  = tid & 31;
    const int wave  = tid >> 5;
    const int waveM = wave >> 1;
    const int waveN = wave & 1;
    const int bm = blockIdx.x * TM;
    const int bn = blockIdx.y * TN;

    v8f acc[2][2] = {};

    const int am  = lane & 15;
    const int ak  = (lane >> 4) * 2;
    const int bnl = lane & 15;

    for (int k0 = 0; k0 < K; k0 += TK) {
        __syncthreads();
        for (int j = tid; j < (TM * TK) / 4; j += 256) {
            int r  = j >> 3;
            int cq = (j & 7) * 4;
            v4f v = *(const v4f*)&X[(size_t)(bm + r) * K + (k0 + cq)];
            sX[r][cq + 0] = v.x; sX[r][cq + 1] = v.y;
            sX[r][cq + 2] = v.z; sX[r][cq + 3] = v.w;
        }
        for (int j = tid; j < (TK * TN) / 4; j += 256) {
            int r  = j >> 4;
            int cq = (j & 15) * 4;
            v4f v = *(const v4f*)&W[(size_t)(k0 + r) * N + (bn + cq)];
            sW[cq + 0][r] = v.x; sW[cq + 1][r] = v.y;
            sW[cq + 2][r] = v.z; sW[cq + 3][r] = v.w;
        }
        __syncthreads();

#pragma unroll
        for (int kk = 0; kk < TK; kk += 4) {
            v2f afr[2];
#pragma unroll
            for (int i = 0; i < 2; ++i) {
                int row = waveM * 32 + i * 16 + am;
                afr[i] = *(const v2f*)&sX[row][kk + ak];
            }
            v2f bfr[2];
#pragma unroll
            for (int j = 0; j < 2; ++j) {
                int col = waveN * 32 + j * 16 + bnl;
                bfr[j] = *(const v2f*)&sW[col][kk + ak];
            }
#pragma unroll
            for (int i = 0; i < 2; ++i)
#pragma unroll
                for (int j = 0; j < 2; ++j)
                    acc[i][j] = __builtin_amdgcn_wmma_f32_16x16x4_f32(
                        false, afr[i], false, bfr[j], (short)0, acc[i][j], false, false);
        }
    }

    const int col0  = lane & 15;
    const int rhalf = (lane >> 4) * 8;
#pragma unroll
    for (int i = 0; i < 2; ++i)
#pragma unroll
        for (int j = 0; j < 2; ++j) {
            int ncol = bn + waveN * 32 + j * 16 + col0;
            float bv = bias[ncol];
#pragma unroll
            for (int r = 0; r < 8; ++r) {
                int row = bm + waveM * 32 + i * 16 + r + rhalf;
                out[(size_t)row * N + ncol] = acc[i][j][r] + bv;
            }
        }
}

// ---------------------------------------------------------------------------
// Chunked parallel scan for h_t = a_t * h_{t-1} + b_t, h_0 = 0.
// Phase 1: per-(b,h,chunk) thread composes the chunk into (Ac, Bc).
// Phase 2: per-(b,h) thread scans chunk summaries -> h at each chunk start.
// Phase 3: per-(b,h,chunk) thread replays chunk from its start value.
// Layouts: a,b,h are [B*T, H]; summaries are [nchunk, B*H] (coalesced).
// ---------------------------------------------------------------------------
__global__ __launch_bounds__(256)
void scan_local(const float* __restrict__ A, const float* __restrict__ Bv,
                float* __restrict__ cA, float* __restrict__ cB,
                int Hc, int T, int nchunk, int BH)
{
    int idx = blockIdx.x * blockDim.x + threadIdx.x;
    int h   = idx % Hc;
    int tmp = idx / Hc;
    int c   = tmp % nchunk;
    int b   = tmp / nchunk;
    size_t base = ((size_t)(b * T + c * CHUNK)) * Hc + h;

    float Ac = 1.0f, Bc = 0.0f;
    for (int t = 0; t < CHUNK; ++t) {
        float a  = A[base + (size_t)t * Hc];
        float bb = Bv[base + (size_t)t * Hc];
        Bc = fmaf(a, Bc, bb);   // combine((Ac,Bc),(a,bb))
        Ac = Ac * a;
    }
    int ch = b * Hc + h;
    cA[(size_t)c * BH + ch] = Ac;
    cB[(size_t)c * BH + ch] = Bc;
}

__global__ __launch_bounds__(256)
void scan_chunks(const float* __restrict__ cA, const float* __restrict__ cB,
                 float* __restrict__ hstart, int BH, int nchunk)
{
    int ch = blockIdx.x * blockDim.x + threadIdx.x;
    if (ch >= BH) return;
    float h = 0.0f;
    for (int c = 0; c < nchunk; ++c) {
        hstart[(size_t)c * BH + ch] = h;
        h = fmaf(cA[(size_t)c * BH + ch], h, cB[(size_t)c * BH + ch]);
    }
}

__global__ __launch_bounds__(256)
void scan_apply(const float* __restrict__ A, const float* __restrict__ Bv,
                const float* __restrict__ hstart, float* __restrict__ Hout,
                int Hc, int T, int nchunk, int BH)
{
    int idx = blockIdx.x * blockDim.x + threadIdx.x;
    int h   = idx % Hc;
    int tmp = idx / Hc;
    int c   = tmp % nchunk;
    int b   = tmp / nchunk;
    int ch  = b * Hc + h;
    size_t base = ((size_t)(b * T + c * CHUNK)) * Hc + h;

    float hv = hstart[(size_t)c * BH + ch];
    for (int t = 0; t < CHUNK; ++t) {
        size_t idx2 = base + (size_t)t * Hc;
        hv = fmaf(A[idx2], hv, Bv[idx2]);
        Hout[idx2] = hv;
    }
}

// ---------------------------------------------------------------------------
extern "C" void kernel_launch(void* const* d_in, const int* in_sizes, int n_in,
                              void* d_out, int out_size, void* d_ws, size_t ws_size,
                              hipStream_t stream)
{
    const float* x   = (const float*)d_in[0];
    const float* Wz0 = (const float*)d_in[1];
    const float* bz0 = (const float*)d_in[2];
    const float* Wh0 = (const float*)d_in[3];
    const float* bh0 = (const float*)d_in[4];
    const float* Wz1 = (const float*)d_in[5];
    const float* bz1 = (const float*)d_in[6];
    const float* Wh1 = (const float*)d_in[7];
    const float* bh1 = (const float*)d_in[8];
    const float* Wfc = (const float*)d_in[9];
    const float* bfc = (const float*)d_in[10];
    float* out = (float*)d_out;

    // Workspace layout (floats)
    const size_t MH = (size_t)MROWS * HDIM;          // 33,554,432
    const int    BH = BSZ * HDIM;                    // 8192
    float* wsA    = (float*)d_ws;                    // a coefficients [M,H]
    float* wsB    = wsA + MH;                        // b coefficients [M,H]
    float* wsH    = wsB + MH;                        // hidden state   [M,H]
    float* wsCA   = wsH + MH;                        // chunk A  [nchunk, BH]
    float* wsCB   = wsCA + (size_t)NCHUNK * BH;      // chunk B
    float* wsHS   = wsCB + (size_t)NCHUNK * BH;      // chunk start h

    dim3 blk(256);
    dim3 gDual(MROWS / TM, HDIM / TN);               // 256 x 16
    dim3 gFC(MROWS / TM, DOUT / TN);                 // 256 x 8
    int  gScan  = (BH * NCHUNK) / 256;               // 1024 blocks
    int  gChunk = (BH + 255) / 256;                  // 32 blocks

    // ---- Layer 0: gates + candidate from x, then scan ----
    gemm_dual_mingru<<<gDual, blk, 0, stream>>>(x, Wz0, bz0, Wh0, bh0,
                                                wsA, wsB, MROWS, DIN, HDIM);
    scan_local <<<gScan,  blk, 0, stream>>>(wsA, wsB, wsCA, wsCB, HDIM, TSEQ, NCHUNK, BH);
    scan_chunks<<<gChunk, blk, 0, stream>>>(wsCA, wsCB, wsHS, BH, NCHUNK);
    scan_apply <<<gScan,  blk, 0, stream>>>(wsA, wsB, wsHS, wsH, HDIM, TSEQ, NCHUNK, BH);

    // ---- Layer 1: gates + candidate from h0, then scan (reuse buffers) ----
    gemm_dual_mingru<<<gDual, blk, 0, stream>>>(wsH, Wz1, bz1, Wh1, bh1,
                                                wsA, wsB, MROWS, HDIM, HDIM);
    scan_local <<<gScan,  blk, 0, stream>>>(wsA, wsB, wsCA, wsCB, HDIM, TSEQ, NCHUNK, BH);
    scan_chunks<<<gChunk, blk, 0, stream>>>(wsCA, wsCB, wsHS, BH, NCHUNK);
    scan_apply <<<gScan,  blk, 0, stream>>>(wsA, wsB, wsHS, wsH, HDIM, TSEQ, NCHUNK, BH);

    // ---- Final FC ----
    gemm_bias<<<gFC, blk, 0, stream>>>(wsH, Wfc, bfc, out, MROWS, HDIM, DOUT);
}